// CrossHybridMemoryMultiFocalPercent_8186207666551
// MI455X (gfx1250) — compile-verified
//
#include <hip/hip_runtime.h>
#include <hip/hip_bf16.h>

// Problem constants (match reference).
#define TEMPR 0.05f
#define TOPP  0.1f
#define EPSF  1e-6f

constexpr int B  = 256;
constexpr int NF = 256;
constexpr int M  = 65536;
constexpr int C  = 8192;

typedef __attribute__((ext_vector_type(2))) float v2f;
typedef __attribute__((ext_vector_type(4))) float v4f;
typedef __attribute__((ext_vector_type(8))) float v8f;

// ---------------------------------------------------------------- utilities
__device__ __forceinline__ float block_sum(float v, float* red) {
    #pragma unroll
    for (int o = 16; o; o >>= 1) v += __shfl_down(v, o, 32);
    const int wid = threadIdx.x >> 5, lane = threadIdx.x & 31;
    __syncthreads();
    if (lane == 0) red[wid] = v;
    __syncthreads();
    float r = 0.f;
    #pragma unroll
    for (int i = 0; i < 8; ++i) r += red[i];
    return r;
}

__device__ __forceinline__ float block_min(float v, float* red) {
    #pragma unroll
    for (int o = 16; o; o >>= 1) v = fminf(v, __shfl_down(v, o, 32));
    const int wid = threadIdx.x >> 5, lane = threadIdx.x & 31;
    __syncthreads();
    if (lane == 0) red[wid] = v;
    __syncthreads();
    float r = 3.4e38f;
    #pragma unroll
    for (int i = 0; i < 8; ++i) r = fminf(r, red[i]);
    return r;
}

__device__ __forceinline__ float block_max(float v, float* red) {
    #pragma unroll
    for (int o = 16; o; o >>= 1) v = fmaxf(v, __shfl_down(v, o, 32));
    const int wid = threadIdx.x >> 5, lane = threadIdx.x & 31;
    __syncthreads();
    if (lane == 0) red[wid] = v;
    __syncthreads();
    float r = -3.4e38f;
    #pragma unroll
    for (int i = 0; i < 8; ++i) r = fmaxf(r, red[i]);
    return r;
}

// ------------------------------------------------------------- kernel: zero
// Vectorized b128 zeroing of Fc | counts (contiguous in workspace).
__global__ void k_zero(float* __restrict__ base, float* __restrict__ out) {
    const long n4 = ((long)C * NF + C) / 4;           // both multiples of 4
    v4f z = {};
    for (long i = blockIdx.x * (long)blockDim.x + threadIdx.x; i < n4;
         i += (long)gridDim.x * blockDim.x)
        *(v4f*)(base + 4 * i) = z;
    if (blockIdx.x == 0 && threadIdx.x == 0) out[0] = 0.f;
}

// -------------------------------------------------- kernel: L2-normalize x
__global__ __launch_bounds__(256) void k_normalize(const float* __restrict__ results,
                                                   float* __restrict__ x) {
    __shared__ float red[8];
    const int b = blockIdx.x, t = threadIdx.x;        // NF == blockDim == 256
    const float v  = results[b * NF + t];
    const float ss = block_sum(v * v, red);
    x[b * NF + t] = v * rsqrtf(ss);
}

// ----------------------------------- kernel: segment-sum features -> class
// float4 loads: 64 lanes cover one 256-float feature row; dominant 64 MB pass.
__global__ void k_accum(const float* __restrict__ features,
                        const int* __restrict__ labels,
                        float* __restrict__ Fc, float* __restrict__ counts) {
    const long n4 = (long)M * NF / 4;                 // 4 floats per thread-step
    for (long i = blockIdx.x * (long)blockDim.x + threadIdx.x; i < n4;
         i += (long)gridDim.x * blockDim.x) {
        const int m = (int)(i >> 6);                  // NF/4 == 64
        const int k = (int)(i & 63) * 4;
        const int lab = labels[m];
        const v4f f = *(const v4f*)(features + (long)m * NF + k);
        float* dst = Fc + (long)lab * NF + k;
        unsafeAtomicAdd(dst + 0, f.x);
        unsafeAtomicAdd(dst + 1, f.y);
        unsafeAtomicAdd(dst + 2, f.z);
        unsafeAtomicAdd(dst + 3, f.w);
        if (k == 0) unsafeAtomicAdd(&counts[lab], 1.0f);
    }
}

// --------------------------- kernel: Fc[c,:] *= 1/(TEMP*count[c]) (mean+T)
__global__ void k_scale(float* __restrict__ Fc, const float* __restrict__ counts) {
    const long n4 = (long)C * NF / 4;
    for (long i = blockIdx.x * (long)blockDim.x + threadIdx.x; i < n4;
         i += (long)gridDim.x * blockDim.x) {
        const int c = (int)(i >> 6);
        const float cnt = counts[c];
        const float s = (cnt > 0.f) ? 1.0f / (TEMPR * cnt) : 0.f;
        v4f f = *(v4f*)(Fc + 4 * i);
        f.x *= s; f.y *= s; f.z *= s; f.w *= s;
        *(v4f*)(Fc + 4 * i) = f;
    }
}

// -------------------------------------- kernel: sim = x @ Fc^T  (WMMA f32)
// grid: (C/128, B/16); block 256 = 8 waves, one 16x16 tile per wave.
__global__ __launch_bounds__(256) void k_gemm(const float* __restrict__ x,
                                              const float* __restrict__ Fc,
                                              float* __restrict__ sim) {
    __shared__ float ldsX[16 * NF];                   // 16 KB A-tile, shared by 8 waves
    const int mBase = blockIdx.y * 16;
    const int wave  = threadIdx.x >> 5;
    const int lane  = threadIdx.x & 31;
    const int cBase = blockIdx.x * 128 + wave * 16;

    for (int i = threadIdx.x; i < 16 * NF; i += 256)
        ldsX[i] = x[(mBase + (i >> 8)) * NF + (i & 255)];
    __syncthreads();

    // A 16x4 f32: lanes 0-15 row m hold K={0,1}; lanes 16-31 hold K={2,3}.
    // B 4x16 f32: mirrored; B[k][n] = Fc[cBase+n][k] (row-major Fc => contiguous k).
    const int mr   = lane & 15;
    const int koff = (lane >> 4) * 2;
    const float* __restrict__ xrow = ldsX + mr * NF;
    const float* __restrict__ frow = Fc + (long)(cBase + mr) * NF;

    v8f acc = {};
    #pragma unroll 8
    for (int k0 = 0; k0 < NF; k0 += 4) {
        v2f a  = *(const v2f*)(xrow + k0 + koff);
        v2f bb = *(const v2f*)(frow + k0 + koff);
        acc = __builtin_amdgcn_wmma_f32_16x16x4_f32(false, a, false, bb,
                                                    (short)0, acc, false, false);
    }

    // C/D layout: VGPR r -> M = r (lanes 0-15) / r+8 (lanes 16-31), N = lane&15.
    const int nOut = cBase + (lane & 15);
    const int mOff = (lane >> 4) * 8;
    #pragma unroll
    for (int r = 0; r < 8; ++r)
        sim[(long)(mBase + mOff + r) * C + nOut] = acc[r];
}

// ---------------- kernel: masked focal-percent softmax + NLL, one block/row
__global__ __launch_bounds__(256) void k_focal(const float* __restrict__ sim,
                                               const float* __restrict__ counts,
                                               const int* __restrict__ indexes,
                                               const int* __restrict__ labels,
                                               float* __restrict__ out) {
    __shared__ float neg[C];                          // 32 KB row cache
    __shared__ float red[8];
    __shared__ float s_pos;
    const int b = blockIdx.x, t = threadIdx.x;
    const int target = labels[indexes[b]];
    if (t == 0) s_pos = 0.f;
    __syncthreads();

    const float* __restrict__ row = sim + (long)b * C;
    float lsum = 0.f, lmax = 0.f;
    // b128 reads of sim row and counts: 8 float4 per thread.
    for (int c4 = t; c4 < C / 4; c4 += 256) {
        const v4f sv = *(const v4f*)(row + 4 * c4);
        const v4f cv = *(const v4f*)(counts + 4 * c4);
        #pragma unroll
        for (int j = 0; j < 4; ++j) {
            const int c = 4 * c4 + j;
            const float s = (j == 0) ? sv.x : (j == 1) ? sv.y : (j == 2) ? sv.z : sv.w;
            const float cn = (j == 0) ? cv.x : (j == 1) ? cv.y : (j == 2) ? cv.z : cv.w;
            float v = (cn > 0.f) ? __expf(s) : 0.f;   // masked_exps
            if (c == target) { s_pos = v; v = 0.f; }  // split out positive
            neg[c] = v;
            lsum += v;
            lmax  = fmaxf(lmax, v);
        }
    }
    const float neg_sum = block_sum(lsum, red);
    const float maxv    = block_max(lmax, red);
    const float tgt     = TOPP * neg_sum;             // target cumsum (unnormalized)

    // Bisection on keep-threshold: S(t)=sum{v>=t} is monotone decreasing.
    float lo = 0.f, hi = maxv;
    for (int it = 0; it < 32; ++it) {
        const float mid = 0.5f * (lo + hi);
        float s = 0.f;
        for (int c = t; c < C; c += 256) { const float v = neg[c]; if (v >= mid) s += v; }
        s = block_sum(s, red);
        if (s >= tgt) lo = mid; else hi = mid;        // block-uniform update
    }

    // Crossing element e1 = smallest kept value; cum at e1 = sum of kept set.
    float lmin = 3.4e38f, s1 = 0.f;
    for (int c = t; c < C; c += 256) {
        const float v = neg[c];
        if (v >= lo) { lmin = fminf(lmin, v); s1 += v; }
    }
    const float e1   = block_min(lmin, red);
    const float cum1 = block_sum(s1, red);
    const float cum0 = cum1 - e1;                     // cumsum one index earlier

    float minval;
    if (fabsf(cum0 - tgt) < fabsf(cum1 - tgt)) {
        // argmin prefers previous index -> threshold is next-larger element
        float lm2 = 3.4e38f;
        for (int c = t; c < C; c += 256) {
            const float v = neg[c];
            if (v >= lo && v > e1) lm2 = fminf(lm2, v);
        }
        minval = block_min(lm2, red);
        if (!(minval <= maxv)) minval = e1;           // degenerate: single kept value
    } else {
        minval = e1;
    }

    // Final masked sum: kept negatives + positive exp.
    float ks = 0.f;
    for (int c = t; c < C; c += 256) { const float v = neg[c]; if (v >= minval) ks += v; }
    const float kept_sum = block_sum(ks, red);
    const float fsum = kept_sum + s_pos;
    const float prob = s_pos / (fsum + EPSF);
    const float loss = -__logf(prob + 1e-6f);
    if (t == 0) unsafeAtomicAdd(out, loss * (1.0f / (float)B));
}

// --------------------------------------------------------------- launcher
extern "C" void kernel_launch(void* const* d_in, const int* in_sizes, int n_in,
                              void* d_out, int out_size, void* d_ws, size_t ws_size,
                              hipStream_t stream) {
    const float* results  = (const float*)d_in[0];    // [B, NF]
    const float* features = (const float*)d_in[1];    // [M, NF]
    const int*   indexes  = (const int*)d_in[2];      // [B]
    const int*   labels   = (const int*)d_in[3];      // [M]
    float*       out      = (float*)d_out;            // scalar loss

    // Workspace carve-up (floats): x | Fc | counts | sim  (~16.3 MB total)
    float* x      = (float*)d_ws;
    float* Fc     = x + (size_t)B * NF;
    float* counts = Fc + (size_t)C * NF;
    float* simbuf = counts + (size_t)C;

    k_zero<<<1024, 256, 0, stream>>>(Fc, out);        // zeros Fc + counts (contiguous)
    k_normalize<<<B, 256, 0, stream>>>(results, x);
    k_accum<<<4096, 256, 0, stream>>>(features, labels, Fc, counts);
    k_scale<<<1024, 256, 0, stream>>>(Fc, counts);
    dim3 g(C / 128, B / 16);
    k_gemm<<<g, 256, 0, stream>>>(x, Fc, simbuf);
    k_focal<<<B, 256, 0, stream>>>(simbuf, counts, indexes, labels, out);
}